// ADMM_41386304864498
// MI455X (gfx1250) — compile-verified
//
#include <hip/hip_runtime.h>

// ---------------------------------------------------------------------------
// ADMM TV-only solver for MI455X (gfx1250).
// B=4096 rows, K=64, M=9, 200 iterations.
// One wave32 owns 16 rows for the whole solve; all state lives in LDS.
// Matmuls use V_WMMA_F32_16X16X4_F32 (fp32 in / fp32 out, matches reference).
// ---------------------------------------------------------------------------

typedef float v2f __attribute__((ext_vector_type(2)));
typedef float v8f __attribute__((ext_vector_type(8)));

static constexpr int   N_ITER    = 200;
static constexpr float GAMMA_TV  = 2.0f;
static constexpr float ALPHA     = 5.0f;
static constexpr float LAMBDA_TV = 1e-4f;

__device__ __forceinline__ float soft_thresh(float v, float t) {
  float a = fmaxf(fabsf(v) - t, 0.0f);
  return copysignf(a, v);   // sign(0)*0 == 0 either way
}

// ---------------------------------------------------------------------------
// Wave-level (16x64) @ (64x64) fp32 matmul with V_WMMA_F32_16X16X4_F32.
//   S  : 16x64 row-major source (LDS)
//   Bp : 64x64 B-matrix in pair-swizzled layout (LDS):
//        Bp[p*128 + 2*n + j] = B[2p + j][n],  p=0..31, n=0..63, j=0,1
//        so the lane fragment {B[4k+2h][n], B[4k+2h+1][n]} is one aligned b64.
//   D  : 16x64 row-major destination (LDS)
// A-frag layout (32-bit 16x4): lanes 0-15 hold K=0,1 in v0,v1; lanes 16-31
// hold K=2,3.  B-frag mirrors with lane = N.  C: VGPR v -> row v + 8*(lane/16),
// col = 16*t + lane%16.  EXEC must be all ones (no divergence here).
// ---------------------------------------------------------------------------
__device__ __forceinline__ void wave_mm(const float* S, const float* Bp,
                                        float* D, int lane) {
  const int hf = lane >> 4;        // K-half within group of 4
  const int lr = lane & 15;        // M (for A) / N (for B,C)
  const float* arow  = S  + lr * 64 + 2 * hf;
  const float* bbase = Bp + hf * 128 + 2 * lr;
  v8f c0 = {}, c1 = {}, c2 = {}, c3 = {};
#pragma unroll
  for (int k = 0; k < 16; ++k) {
    v2f a = *(const v2f*)(arow + 4 * k);
    const float* bp = bbase + k * 256;        // pair p = 2k + hf
    v2f b0 = *(const v2f*)(bp);
    v2f b1 = *(const v2f*)(bp + 32);
    v2f b2 = *(const v2f*)(bp + 64);
    v2f b3 = *(const v2f*)(bp + 96);
    c0 = __builtin_amdgcn_wmma_f32_16x16x4_f32(false, a, false, b0, (short)0, c0, false, false);
    c1 = __builtin_amdgcn_wmma_f32_16x16x4_f32(false, a, false, b1, (short)0, c1, false, false);
    c2 = __builtin_amdgcn_wmma_f32_16x16x4_f32(false, a, false, b2, (short)0, c2, false, false);
    c3 = __builtin_amdgcn_wmma_f32_16x16x4_f32(false, a, false, b3, (short)0, c3, false, false);
  }
  float* drow = D + (hf * 8) * 64 + lr;
#pragma unroll
  for (int v = 0; v < 8; ++v) {
    drow[v * 64 +  0] = c0[v];
    drow[v * 64 + 16] = c1[v];
    drow[v * 64 + 32] = c2[v];
    drow[v * 64 + 48] = c3[v];
  }
}

// ---------------------------------------------------------------------------
// prep_kernel: single workgroup.
//   G = A^T A  (64x64)
//   Minv = inv(G + GAMMA_TV*DtD + ALPHA*I)  via Gauss-Jordan (SPD, no pivoting)
//   DtD is the circulant Laplacian: 2 on diag, -1 on (i±1 mod 64).
// Writes to workspace, already pair-swizzled for WMMA B-fragments:
//   wsW[p*128 + 2n + j] = Minv[n][2p+j]   (B for x = resid @ inv^T)
//   wsG[p*128 + 2n + j] = G[2p+j][n]      (B for ATb = target @ G)
// Dynamic LDS: G(4096) | aug(64x128=8192) | fac(64)
// ---------------------------------------------------------------------------
__global__ __launch_bounds__(128)
void prep_kernel(const float* __restrict__ A, float* __restrict__ wsW,
                 float* __restrict__ wsG) {
  extern __shared__ float l[];
  float* G   = l;
  float* aug = l + 4096;
  float* fac = l + 4096 + 8192;
  const int tid = threadIdx.x;

  // G = A^T A  (M=9 rows of A, K=64)
  for (int q = tid; q < 4096; q += 128) {
    int i = q >> 6, j = q & 63;
    float s = 0.0f;
#pragma unroll
    for (int m = 0; m < 9; ++m) s += A[m * 64 + i] * A[m * 64 + j];
    G[q] = s;
  }
  __syncthreads();

  // aug = [ G + 2*DtD + 5*I  |  I ]
  for (int q = tid; q < 64 * 128; q += 128) {
    int r = q >> 7, c = q & 127;
    float v;
    if (c < 64) {
      float dtd = (c == r) ? 2.0f
                : ((c == ((r + 1) & 63)) || (c == ((r + 63) & 63))) ? -1.0f : 0.0f;
      v = G[r * 64 + c] + GAMMA_TV * dtd + ((c == r) ? ALPHA : 0.0f);
    } else {
      v = ((c - 64) == r) ? 1.0f : 0.0f;
    }
    aug[q] = v;
  }
  __syncthreads();

  // Gauss-Jordan; thread tid owns column tid (128 columns).
  for (int k = 0; k < 64; ++k) {
    float pinv = 1.0f / aug[k * 128 + k];
    __syncthreads();
    aug[k * 128 + tid] *= pinv;
    __syncthreads();
    if (tid < 64) fac[tid] = (tid == k) ? 0.0f : aug[tid * 128 + k];
    __syncthreads();
    for (int r = 0; r < 64; ++r)
      aug[r * 128 + tid] -= fac[r] * aug[k * 128 + tid];
    __syncthreads();
  }

  // Emit pair-swizzled B matrices.
  for (int q = tid; q < 4096; q += 128) {
    int p = q >> 6, n = q & 63;
    wsW[p * 128 + 2 * n]     = aug[n * 128 + 64 + 2 * p];
    wsW[p * 128 + 2 * n + 1] = aug[n * 128 + 64 + 2 * p + 1];
    wsG[p * 128 + 2 * n]     = G[(2 * p) * 64 + n];
    wsG[p * 128 + 2 * n + 1] = G[(2 * p + 1) * 64 + n];
  }
}

// ---------------------------------------------------------------------------
// admm_kernel: one wave32 per block, 16 rows per block, 256 blocks.
// Dynamic LDS (floats):
//   Wp [0,4096) | xs [4096,5120) | ab [5120,6144) | sc [6144,7168)
//   us [7168,8192) | wv [8192,9216) | es [9216,10240) | ta [10240,11264)
//   Gp overlays [7168,11264) during init only (us/wv/es/ta not yet live).
// Total 11264 floats = 44 KB.
// ---------------------------------------------------------------------------
__global__ __launch_bounds__(32)
void admm_kernel(const float* __restrict__ target, const float* __restrict__ x0,
                 const float* __restrict__ wsW, const float* __restrict__ wsG,
                 float* __restrict__ out) {
  extern __shared__ float lds[];
  float* Wp = lds;            // 4096: inv^T in B-pair layout
  float* xs = lds + 4096;     // x        (16x64)
  float* ab = lds + 5120;     // ATb
  float* sc = lds + 6144;     // resid / target scratch
  float* us = lds + 7168;     // u_tv
  float* wv = lds + 8192;     // w
  float* es = lds + 9216;     // eta_tv
  float* ta = lds + 10240;    // tau
  float* Gp = lds + 7168;     // init-time overlay (4096 floats)

  const int lane = threadIdx.x;          // 32 threads = 1 wave
  const int r0 = blockIdx.x * 16;        // global row base

  // ---- init loads -------------------------------------------------------
  for (int q = lane; q < 4096; q += 32) Wp[q] = wsW[q];
  for (int q = lane; q < 4096; q += 32) Gp[q] = wsG[q];
  for (int q = lane; q < 1024; q += 32) {
    xs[q] = x0[r0 * 64 + q];
    sc[q] = target[r0 * 64 + q];
  }
  __syncthreads();

  // ATb = target_tile @ G
  wave_mm(sc, Gp, ab, lane);
  __builtin_amdgcn_wave_barrier();

  // eta = tau = 0  (safe now: Gp no longer needed; same-wave DS is in-order)
  for (int q = lane; q < 1024; q += 32) { es[q] = 0.0f; ta[q] = 0.0f; }
  __builtin_amdgcn_wave_barrier();

  const float inv_g = 1.0f / GAMMA_TV;          // 0.5
  const float thr   = LAMBDA_TV / GAMMA_TV;     // 5e-5
  const float inv_a = 1.0f / ALPHA;             // 0.2

  // ---- 200 ADMM iterations ---------------------------------------------
  for (int it = 0; it < N_ITER; ++it) {
    // phase 1: u = soft(D(x) + eta/g, lam/g);  w = max(x + tau/a, 0)
    for (int e = 0; e < 32; ++e) {
      int q = e * 32 + lane;
      int m = q >> 6, i = q & 63, i1 = (i + 1) & 63;
      float xv = xs[m * 64 + i];
      float dv = xs[m * 64 + i1] - xv;
      us[q] = soft_thresh(dv + inv_g * es[q], thr);
      wv[q] = fmaxf(xv + inv_a * ta[q], 0.0f);
    }
    __builtin_amdgcn_wave_barrier();

    // phase 2: resid = ATb + D(g*u - eta) + a*w - tau
    for (int e = 0; e < 32; ++e) {
      int q = e * 32 + lane;
      int m = q >> 6, i = q & 63, i1 = (i + 1) & 63;
      float z  = GAMMA_TV * us[q]             - es[q];
      float z1 = GAMMA_TV * us[m * 64 + i1]   - es[m * 64 + i1];
      sc[q] = ab[q] + (z1 - z) + ALPHA * wv[q] - ta[q];
    }
    __builtin_amdgcn_wave_barrier();

    // phase 3: x = resid @ inv^T   (64 x V_WMMA_F32_16X16X4_F32)
    wave_mm(sc, Wp, xs, lane);
    __builtin_amdgcn_wave_barrier();

    // phase 4: eta += g*(D(x) - u);  tau += a*(x - w)
    for (int e = 0; e < 32; ++e) {
      int q = e * 32 + lane;
      int m = q >> 6, i = q & 63, i1 = (i + 1) & 63;
      float xv = xs[q];
      float dv = xs[m * 64 + i1] - xv;
      es[q] += GAMMA_TV * (dv - us[q]);
      ta[q] += ALPHA * (xv - wv[q]);
    }
    __builtin_amdgcn_wave_barrier();
  }

  // ---- writeback --------------------------------------------------------
  for (int q = lane; q < 1024; q += 32)
    out[r0 * 64 + q] = xs[q];
}

// ---------------------------------------------------------------------------
extern "C" void kernel_launch(void* const* d_in, const int* in_sizes, int n_in,
                              void* d_out, int out_size, void* d_ws, size_t ws_size,
                              hipStream_t stream) {
  const float* target = (const float*)d_in[0];  // (4096, 64)
  const float* A      = (const float*)d_in[1];  // (9, 64)
  const float* x0     = (const float*)d_in[2];  // (4096, 64)
  float* out = (float*)d_out;                   // (4096, 64)

  float* wsW = (float*)d_ws;                    // 4096 floats
  float* wsG = wsW + 4096;                      // 4096 floats (32 KB total)

  // one-time 64x64 inverse + Gram matrix, pre-swizzled for WMMA B-fragments
  prep_kernel<<<1, 128, (4096 + 8192 + 64) * sizeof(float), stream>>>(A, wsW, wsG);

  // 4096 rows / 16 rows-per-wave = 256 single-wave workgroups, 44 KB LDS each
  admm_kernel<<<256, 32, 11264 * sizeof(float), stream>>>(target, x0, wsW, wsG, out);
}